// GAT_3951369912452
// MI455X (gfx1250) — compile-verified
//
#include <hip/hip_runtime.h>
#include <math.h>

#define GAT_N 50000
#define GAT_E 800000
#define GAT_H 4
#define GAT_SLOPE 0.2f

typedef __attribute__((ext_vector_type(2))) float v2f;
typedef __attribute__((ext_vector_type(8))) float v8f;

// ---------------------------------------------------------------------------
// fp32 WMMA GEMM: C[M,NCOL] = A[M,K] @ B[K,NCOL]  (row-major everywhere)
// One wave -> one 32x16 C tile (two 16x16 accumulators sharing the B fetch),
// K stepped by 4 via V_WMMA_F32_16X16X4_F32. K/NCOL are compile-time so all
// strides become immediate offsets in the inner loop.
// A layout  (16x4): lane L holds (m=L%16, k = 2*(L/16) + vgpr)   -> v2f
// B layout  (4x16): lane L holds (k = 2*(L/16) + vgpr, n=L%16)   -> v2f
// C/D layout(16x16): lane L, vgpr r holds (m = r + 8*(L/16), n=L%16)
// ---------------------------------------------------------------------------
template <int K, int NCOL>
__global__ __launch_bounds__(128) void gat_gemm_wmma(
    const float* __restrict__ A, const float* __restrict__ B,
    float* __restrict__ C, int M)
{
  const int wave = threadIdx.x >> 5;
  const int lane = threadIdx.x & 31;
  const int row0 = blockIdx.x << 5;                    // 32 rows per block-row
  const int col0 = ((blockIdx.y << 2) + wave) << 4;
  if (col0 >= NCOL) return;

  const int m     = lane & 15;
  const int khalf = (lane >> 4) << 1;                  // 0 or 2
  const int n     = col0 + m;
  const bool hi_valid = (row0 + 16) < M;               // tiles are 16-aligned vs M

  const float* arow0 = A + (size_t)(row0 + m) * K + khalf;
  const float* arow1 = hi_valid ? (arow0 + (size_t)16 * K) : arow0;
  const float* bcol  = B + (size_t)khalf * NCOL + n;

  v8f acc0 = {}, acc1 = {};
#pragma unroll 8
  for (int k0 = 0; k0 < K; k0 += 4) {
    v2f a0 = *(const v2f*)arow0;                       // A[row   ][k0+khalf..+1]
    v2f a1 = *(const v2f*)arow1;                       // A[row+16][k0+khalf..+1]
    v2f b;
    b.x = bcol[0];                                     // B[k0+khalf  ][n]
    b.y = bcol[NCOL];                                  // B[k0+khalf+1][n]
    acc0 = __builtin_amdgcn_wmma_f32_16x16x4_f32(
        false, a0, false, b, (short)0, acc0, false, false);
    acc1 = __builtin_amdgcn_wmma_f32_16x16x4_f32(
        false, a1, false, b, (short)0, acc1, false, false);
    arow0 += 4;
    arow1 += 4;
    bcol  += NCOL * 4;
  }

  const int rsub = (lane >> 4) << 3;
  float* crow0 = C + (size_t)(row0 + rsub) * NCOL + n;
#pragma unroll
  for (int r = 0; r < 8; ++r) crow0[(size_t)r * NCOL] = acc0[r];
  if (hi_valid) {
    float* crow1 = C + (size_t)(row0 + 16 + rsub) * NCOL + n;
#pragma unroll
    for (int r = 0; r < 8; ++r) crow1[(size_t)r * NCOL] = acc1[r];
  }
}

// ---------------------------------------------------------------------------
// el[n,h] = <feat[n, h*D : h*D+D], al[h]>,  er likewise.
// ---------------------------------------------------------------------------
__global__ void gat_attn_scores(const float* __restrict__ feat,
                                const float* __restrict__ al,
                                const float* __restrict__ ar,
                                float* __restrict__ el, float* __restrict__ er,
                                int F, int D)
{
  int idx = blockIdx.x * blockDim.x + threadIdx.x;   // n*H + h
  if (idx >= GAT_N * GAT_H) return;
  int nd = idx >> 2, h = idx & 3;
  const float* f   = feat + (size_t)nd * F + h * D;
  const float* alh = al + h * D;
  const float* arh = ar + h * D;
  float sl = 0.f, sr = 0.f;
  for (int d = 0; d < D; ++d) {
    float v = f[d];
    sl = fmaf(v, alh[d], sl);
    sr = fmaf(v, arh[d], sr);
  }
  el[idx] = sl;
  er[idx] = sr;
}

__global__ void gat_init_mden(float* __restrict__ mx, float* __restrict__ den, int count)
{
  int idx = blockIdx.x * blockDim.x + threadIdx.x;
  if (idx < count) {
    mx[idx]  = __int_as_float((int)0xFF800000u);  // -inf, exact bits for atomic trick
    den[idx] = 0.f;
  }
}

__device__ __forceinline__ float gat_leaky(float x) {
  return x > 0.f ? x : GAT_SLOPE * x;
}

// float atomic-max via monotone int/uint ordering trick
__device__ __forceinline__ void gat_atomic_max_f32(float* addr, float v) {
  if (v >= 0.f) atomicMax((int*)addr, __float_as_int(v));
  else          atomicMin((unsigned int*)addr, __float_as_uint(v));
}

__global__ void gat_edge_max(const int* __restrict__ src, const int* __restrict__ dst,
                             const float* __restrict__ el, const float* __restrict__ er,
                             float* __restrict__ mx)
{
  int idx = blockIdx.x * blockDim.x + threadIdx.x;   // e*H + h
  if (idx >= GAT_E * GAT_H) return;
  int e = idx >> 2, h = idx & 3;
  int s = src[e], d = dst[e];
  float x = gat_leaky(el[s * GAT_H + h] + er[d * GAT_H + h]);
  gat_atomic_max_f32(&mx[d * GAT_H + h], x);
}

__global__ void gat_edge_exp(const int* __restrict__ src, const int* __restrict__ dst,
                             const float* __restrict__ el, const float* __restrict__ er,
                             const float* __restrict__ mx,
                             float* __restrict__ ex, float* __restrict__ den)
{
  int idx = blockIdx.x * blockDim.x + threadIdx.x;
  if (idx >= GAT_E * GAT_H) return;
  int e = idx >> 2, h = idx & 3;
  int s = src[e], d = dst[e];
  float x = gat_leaky(el[s * GAT_H + h] + er[d * GAT_H + h]);
  float v = expf(x - mx[d * GAT_H + h]);
  ex[idx] = v;
  atomicAdd(&den[d * GAT_H + h], v);
}

// one wave per edge; lane-strided coalesced atomic adds over F features
__global__ __launch_bounds__(256) void gat_edge_aggregate(
    const int* __restrict__ src, const int* __restrict__ dst,
    const float* __restrict__ feat, const float* __restrict__ ex,
    const float* __restrict__ den, float* __restrict__ out, int F, int D)
{
  int e = blockIdx.x * (blockDim.x >> 5) + (threadIdx.x >> 5);
  if (e >= GAT_E) return;
  int lane = threadIdx.x & 31;
  int s = src[e], d = dst[e];
  const float* fs = feat + (size_t)s * F;
  float*       od = out  + (size_t)d * F;
  __builtin_prefetch(fs + lane, 0, 0);
  float alpha[GAT_H];
#pragma unroll
  for (int h = 0; h < GAT_H; ++h)
    alpha[h] = ex[(size_t)e * GAT_H + h] / den[(size_t)d * GAT_H + h];
  for (int f = lane; f < F; f += 32)
    atomicAdd(&od[f], fs[f] * alpha[(unsigned)f / (unsigned)D]);
}

__global__ void gat_finalize_elu(float* __restrict__ acc, const float* __restrict__ bias,
                                 const float* __restrict__ res, int total)
{
  int idx = blockIdx.x * blockDim.x + threadIdx.x;
  if (idx >= total) return;
  float v = acc[idx] + bias[idx & 255];              // F == 256 for layers 0/1
  if (res) v += res[idx];
  acc[idx] = v > 0.f ? v : (expf(v) - 1.f);
}

__global__ void gat_finalize_out(const float* __restrict__ acc, const float* __restrict__ bias,
                                 const float* __restrict__ res2, float* __restrict__ out)
{
  const int C = 40;
  int idx = blockIdx.x * blockDim.x + threadIdx.x;   // n*C + c
  if (idx >= GAT_N * C) return;
  int n = idx / C, c = idx % C;
  float sum = 0.f;
#pragma unroll
  for (int h = 0; h < GAT_H; ++h) {
    int j = n * (GAT_H * C) + h * C + c;
    sum += acc[j] + bias[h * C + c] + res2[j];
  }
  out[idx] = sum * 0.25f;                            // mean over heads
}

// ---------------------------------------------------------------------------
extern "C" void kernel_launch(void* const* d_in, const int* in_sizes, int n_in,
                              void* d_out, int out_size, void* d_ws, size_t ws_size,
                              hipStream_t stream)
{
  const float* x     = (const float*)d_in[0];
  const int*   src   = (const int*)d_in[1];
  const int*   dst   = (const int*)d_in[2];
  const float* W0    = (const float*)d_in[3];
  const float* al0   = (const float*)d_in[4];
  const float* ar0   = (const float*)d_in[5];
  const float* b0    = (const float*)d_in[6];
  const float* W1    = (const float*)d_in[7];
  const float* al1   = (const float*)d_in[8];
  const float* ar1   = (const float*)d_in[9];
  const float* b1    = (const float*)d_in[10];
  const float* W2    = (const float*)d_in[11];
  const float* al2   = (const float*)d_in[12];
  const float* ar2   = (const float*)d_in[13];
  const float* b2    = (const float*)d_in[14];
  const float* Wres2 = (const float*)d_in[15];

  const size_t NF = (size_t)GAT_N * 256;   // full-width feature buffer
  const size_t NG = (size_t)GAT_N * 160;   // layer-2 width
  float* ws   = (float*)d_ws;
  float* bufA = ws;                        // feats of current layer
  float* bufB = bufA + NF;                 // h0 / res2
  float* bufC = bufB + NF;                 // h1
  float* bufD = bufC + NF;                 // layer-2 aggregation
  float* EX   = bufD + NG;                 // per-edge exp  (E*H)
  float* EL   = EX + (size_t)GAT_E * GAT_H;
  float* ER   = EL + (size_t)GAT_N * GAT_H;
  float* MX   = ER + (size_t)GAT_N * GAT_H;
  float* DEN  = MX + (size_t)GAT_N * GAT_H;

  dim3 gemmBlk(128);
  const int rowTiles = (GAT_N + 31) / 32;  // 32-row macro tiles
  dim3 gemmGrid256(rowTiles, 4);           // 16 col tiles / 4 waves
  dim3 gemmGrid160(rowTiles, 3);           // 10 col tiles, guarded

  const int ehThreads = GAT_E * GAT_H;
  const int ehBlocks  = (ehThreads + 255) / 256;
  const int nhThreads = GAT_N * GAT_H;
  const int nhBlocks  = (nhThreads + 255) / 256;
  const int aggBlocks = (GAT_E + 7) / 8;   // 8 waves per block, wave per edge
  const int nfBlocks  = (int)((NF + 255) / 256);

  // ---------------- Layer 0: x @ W0, softmax-aggregate, +b0, ELU ----------
  gat_gemm_wmma<256, 256><<<gemmGrid256, gemmBlk, 0, stream>>>(x, W0, bufA, GAT_N);
  gat_attn_scores<<<nhBlocks, 256, 0, stream>>>(bufA, al0, ar0, EL, ER, 256, 64);
  gat_init_mden<<<nhBlocks, 256, 0, stream>>>(MX, DEN, nhThreads);
  gat_edge_max<<<ehBlocks, 256, 0, stream>>>(src, dst, EL, ER, MX);
  gat_edge_exp<<<ehBlocks, 256, 0, stream>>>(src, dst, EL, ER, MX, EX, DEN);
  (void)hipMemsetAsync(bufB, 0, NF * sizeof(float), stream);
  gat_edge_aggregate<<<aggBlocks, 256, 0, stream>>>(src, dst, bufA, EX, DEN, bufB, 256, 64);
  gat_finalize_elu<<<nfBlocks, 256, 0, stream>>>(bufB, b0, nullptr, (int)NF);

  // ---------------- Layer 1: h0 @ W1, + identity residual, ELU ------------
  gat_gemm_wmma<256, 256><<<gemmGrid256, gemmBlk, 0, stream>>>(bufB, W1, bufA, GAT_N);
  gat_attn_scores<<<nhBlocks, 256, 0, stream>>>(bufA, al1, ar1, EL, ER, 256, 64);
  gat_init_mden<<<nhBlocks, 256, 0, stream>>>(MX, DEN, nhThreads);
  gat_edge_max<<<ehBlocks, 256, 0, stream>>>(src, dst, EL, ER, MX);
  gat_edge_exp<<<ehBlocks, 256, 0, stream>>>(src, dst, EL, ER, MX, EX, DEN);
  (void)hipMemsetAsync(bufC, 0, NF * sizeof(float), stream);
  gat_edge_aggregate<<<aggBlocks, 256, 0, stream>>>(src, dst, bufA, EX, DEN, bufC, 256, 64);
  gat_finalize_elu<<<nfBlocks, 256, 0, stream>>>(bufC, b1, bufB, (int)NF);

  // ---------------- Layer 2: h1 @ W2 (+ h1 @ Wres2), head mean ------------
  gat_gemm_wmma<256, 160><<<gemmGrid160, gemmBlk, 0, stream>>>(bufC, W2, bufA, GAT_N);
  gat_gemm_wmma<256, 160><<<gemmGrid160, gemmBlk, 0, stream>>>(bufC, Wres2, bufB, GAT_N);
  gat_attn_scores<<<nhBlocks, 256, 0, stream>>>(bufA, al2, ar2, EL, ER, 160, 40);
  gat_init_mden<<<nhBlocks, 256, 0, stream>>>(MX, DEN, nhThreads);
  gat_edge_max<<<ehBlocks, 256, 0, stream>>>(src, dst, EL, ER, MX);
  gat_edge_exp<<<ehBlocks, 256, 0, stream>>>(src, dst, EL, ER, MX, EX, DEN);
  (void)hipMemsetAsync(bufD, 0, NG * sizeof(float), stream);
  gat_edge_aggregate<<<aggBlocks, 256, 0, stream>>>(src, dst, bufA, EX, DEN, bufD, 160, 40);
  gat_finalize_out<<<(GAT_N * 40 + 255) / 256, 256, 0, stream>>>(bufD, b2, bufB, (float*)d_out);
}